// CTANEmbedding_41171556500204
// MI455X (gfx1250) — compile-verified
//
#include <hip/hip_runtime.h>
#include <math.h>

#define NN 50000
#define NE 800000
#define D 128
#define K_ENC 256
#define K_EDGE 160
#define TDIM 32
#define NUM_ITERS 3
#define EPSILONF 0.1f
#define GAMMAF 0.1f

typedef __attribute__((ext_vector_type(2))) float v2f;
typedef __attribute__((ext_vector_type(8))) float v8f;

// -------- order-preserving float<->uint map for atomic segment-max --------
__device__ __forceinline__ unsigned f2ord(float f) {
  unsigned i = __float_as_uint(f);
  return (i & 0x80000000u) ? ~i : (i | 0x80000000u);
}
__device__ __forceinline__ float ord2f(unsigned u) {
  unsigned i = (u & 0x80000000u) ? (u ^ 0x80000000u) : ~u;
  return __uint_as_float(i);
}

// =====================================================================
// WMMA fp32 16x16x4 tile GEMM: C(16x16) += A(16xK) * B(KxN=16)
// A staged in LDS row-major [16][lda]; B[k][n] = W[n*K + k] (i.e. h @ W^T)
// Layouts per CDNA5 ISA 7.12.2:
//   A frag: lanes 0-15 m=lane, v0=K(k0), v1=K(k0+1); lanes 16-31 K(k0+2/3)
//   B frag: lanes 0-15 n=lane, v0=K(k0), v1=K(k0+1); lanes 16-31 K(k0+2/3)
//   C/D  : vgpr r, lanes 0-15 -> (m=r, n=lane); lanes 16-31 -> (m=r+8)
// =====================================================================
__device__ __forceinline__ v8f wmma_f32_tile(const float* __restrict__ sA, int lda,
                                             const float* __restrict__ Wn, int K,
                                             int lane) {
  const int half = lane >> 4;
  const int lm = lane & 15;
  v8f c = {};
  for (int k0 = 0; k0 < K; k0 += 4) {
    v2f a, b;
    a.x = sA[lm * lda + k0 + 2 * half];
    a.y = sA[lm * lda + k0 + 2 * half + 1];
    b.x = Wn[lm * K + k0 + 2 * half];
    b.y = Wn[lm * K + k0 + 2 * half + 1];
    c = __builtin_amdgcn_wmma_f32_16x16x4_f32(false, a, false, b, (short)0, c,
                                              false, false);
  }
  return c;
}

// ---------------------------------------------------------------------
// Edge features: e = [msg | cos(|last_update[src]-t| * te_w + te_b)] @ We^T + be
// one block = 16 edges, 4 waves each handle 2 N-tiles of 16
// ---------------------------------------------------------------------
__global__ __launch_bounds__(128) void k_edgefeat(
    const float* __restrict__ msg, const float* __restrict__ tarr,
    const float* __restrict__ last_update, const int* __restrict__ src,
    const float* __restrict__ te_w, const float* __restrict__ te_b,
    const float* __restrict__ We, const float* __restrict__ be,
    float* __restrict__ efeat) {
  const int lda = 161;  // odd stride: LDS bank-conflict free
  __shared__ float sA[16 * 161];
  __shared__ float sRel[16];
  const int e0 = blockIdx.x * 16;
  const int tid = threadIdx.x;
  if (tid < 16) {
    int e = e0 + tid;
    sRel[tid] = fabsf(last_update[src[e]] - tarr[e]);
  }
  __syncthreads();
  for (int idx = tid; idx < 16 * K_EDGE; idx += 128) {
    int r = idx / K_EDGE;
    int c = idx - r * K_EDGE;
    float val;
    if (c < D) {
      val = msg[(size_t)(e0 + r) * D + c];
    } else {
      int tc = c - D;
      val = cosf(sRel[r] * te_w[tc] + te_b[tc]);
    }
    sA[r * lda + c] = val;
  }
  __syncthreads();
  const int wave = tid >> 5, lane = tid & 31;
  const int half = lane >> 4, lm = lane & 15;
  for (int nt = 0; nt < 2; ++nt) {
    int n0 = (wave * 2 + nt) * 16;
    v8f c = wmma_f32_tile(sA, lda, We + (size_t)n0 * K_EDGE, K_EDGE, lane);
    float bias = be[n0 + lm];
    for (int r = 0; r < 8; ++r) {
      int m = e0 + r + 8 * half;
      efeat[(size_t)m * D + n0 + lm] = c[r] + bias;
    }
  }
}

// ---------------------------------------------------------------------
// h = x @ enc_W^T + enc_b   (K = 256)
// ---------------------------------------------------------------------
__global__ __launch_bounds__(128) void k_encode(const float* __restrict__ x,
                                                const float* __restrict__ W,
                                                const float* __restrict__ b,
                                                float* __restrict__ h) {
  const int lda = 257;
  __shared__ float sA[16 * 257];
  const int r0 = blockIdx.x * 16;
  const int tid = threadIdx.x;
  for (int idx = tid; idx < 16 * K_ENC; idx += 128) {
    int r = idx >> 8;
    int c = idx & 255;
    sA[r * lda + c] = x[(size_t)(r0 + r) * K_ENC + c];
  }
  __syncthreads();
  const int wave = tid >> 5, lane = tid & 31;
  const int half = lane >> 4, lm = lane & 15;
  for (int nt = 0; nt < 2; ++nt) {
    int n0 = (wave * 2 + nt) * 16;
    v8f c = wmma_f32_tile(sA, lda, W + (size_t)n0 * K_ENC, K_ENC, lane);
    float bias = b[n0 + lm];
    for (int r = 0; r < 8; ++r) {
      int m = r0 + r + 8 * half;
      h[(size_t)m * D + n0 + lm] = c[r] + bias;
    }
  }
}

// ---------------------------------------------------------------------
// q/k/v = h @ W*^T + b*   (shared LDS h tile, K = 128)
// ---------------------------------------------------------------------
__global__ __launch_bounds__(128) void k_qkv(
    const float* __restrict__ h, const float* __restrict__ Wq,
    const float* __restrict__ bq, const float* __restrict__ Wk,
    const float* __restrict__ bk, const float* __restrict__ Wv,
    const float* __restrict__ bv, float* __restrict__ q, float* __restrict__ k,
    float* __restrict__ v) {
  const int lda = 129;
  __shared__ float sA[16 * 129];
  const int r0 = blockIdx.x * 16;
  const int tid = threadIdx.x;
  for (int idx = tid; idx < 16 * D; idx += 128) {
    int r = idx >> 7;
    int c = idx & 127;
    sA[r * lda + c] = h[(size_t)(r0 + r) * D + c];
  }
  __syncthreads();
  const int wave = tid >> 5, lane = tid & 31;
  const int half = lane >> 4, lm = lane & 15;
  const float* Ws[3] = {Wq, Wk, Wv};
  const float* bs[3] = {bq, bk, bv};
  float* outs[3] = {q, k, v};
  for (int m = 0; m < 3; ++m) {
    for (int nt = 0; nt < 2; ++nt) {
      int n0 = (wave * 2 + nt) * 16;
      v8f c = wmma_f32_tile(sA, lda, Ws[m] + (size_t)n0 * D, D, lane);
      float bias = bs[m][n0 + lm];
      for (int r = 0; r < 8; ++r) {
        int row = r0 + r + 8 * half;
        outs[m][(size_t)row * D + n0 + lm] = c[r] + bias;
      }
    }
  }
}

// ---------------------------------------------------------------------
// logits[e] = dot(q[dst], k[src]+efeat[e]) / sqrt(D);  atomic seg-max
// one wave32 per edge, float4 per lane (32*4 = 128)
// ---------------------------------------------------------------------
__global__ __launch_bounds__(256) void k_logits(
    const float* __restrict__ q, const float* __restrict__ kk,
    const float* __restrict__ efeat, const int* __restrict__ src,
    const int* __restrict__ dst, float* __restrict__ logits,
    unsigned* __restrict__ segmax) {
  int wid = (blockIdx.x * blockDim.x + threadIdx.x) >> 5;
  int lane = threadIdx.x & 31;
  if (wid >= NE) return;
  int s = src[wid], d = dst[wid];
  float4 a = ((const float4*)(q + (size_t)d * D))[lane];
  float4 b = ((const float4*)(kk + (size_t)s * D))[lane];
  float4 e = ((const float4*)(efeat + (size_t)wid * D))[lane];
  float sum = a.x * (b.x + e.x) + a.y * (b.y + e.y) + a.z * (b.z + e.z) +
              a.w * (b.w + e.w);
  sum += __shfl_xor(sum, 16);
  sum += __shfl_xor(sum, 8);
  sum += __shfl_xor(sum, 4);
  sum += __shfl_xor(sum, 2);
  sum += __shfl_xor(sum, 1);
  if (lane == 0) {
    float lg = sum * 0.08838834764831845f;  // 1/sqrt(128)
    logits[wid] = lg;
    atomicMax(segmax + d, f2ord(lg));
  }
}

// ex = exp(logit - segmax[dst]); atomic seg-sum (in-place over logits buf)
__global__ __launch_bounds__(256) void k_exp(float* __restrict__ lg,
                                             const int* __restrict__ dst,
                                             const unsigned* __restrict__ segmax,
                                             float* __restrict__ segsum) {
  int e = blockIdx.x * blockDim.x + threadIdx.x;
  if (e >= NE) return;
  int d = dst[e];
  float ex = expf(lg[e] - ord2f(segmax[d]));
  lg[e] = ex;
  atomicAdd(segsum + d, ex);
}

// phi[dst] += alpha * (v[src] + efeat[e]);  one wave32 per edge
__global__ __launch_bounds__(256) void k_scatter(
    const float* __restrict__ v, const float* __restrict__ efeat,
    const int* __restrict__ src, const int* __restrict__ dst,
    const float* __restrict__ ex, const float* __restrict__ segsum,
    float* __restrict__ phi) {
  int wid = (blockIdx.x * blockDim.x + threadIdx.x) >> 5;
  int lane = threadIdx.x & 31;
  if (wid >= NE) return;
  int s = src[wid], d = dst[wid];
  float alpha = ex[wid] / segsum[d];
  float4 a = ((const float4*)(v + (size_t)s * D))[lane];
  float4 b = ((const float4*)(efeat + (size_t)wid * D))[lane];
  float* p = phi + (size_t)d * D + lane * 4;
  atomicAdd(p + 0, (a.x + b.x) * alpha);
  atomicAdd(p + 1, (a.y + b.y) * alpha);
  atomicAdd(p + 2, (a.z + b.z) * alpha);
  atomicAdd(p + 3, (a.w + b.w) * alpha);
}

// ---------------------------------------------------------------------
// h_out = h + eps * tanh(h @ A^T + phi + b),  A = W - W^T - gamma*I
// B built on the fly: B[k][n] = A[n][k] = W[n*D+k] - W[k*D+n] - gamma*(n==k)
// ---------------------------------------------------------------------
__global__ __launch_bounds__(128) void k_update(const float* __restrict__ hin,
                                                const float* __restrict__ Wa,
                                                const float* __restrict__ ba,
                                                const float* __restrict__ phi,
                                                float* __restrict__ hout) {
  const int lda = 129;
  __shared__ float sA[16 * 129];
  const int r0 = blockIdx.x * 16;
  const int tid = threadIdx.x;
  for (int idx = tid; idx < 16 * D; idx += 128) {
    int r = idx >> 7;
    int c = idx & 127;
    sA[r * lda + c] = hin[(size_t)(r0 + r) * D + c];
  }
  __syncthreads();
  const int wave = tid >> 5, lane = tid & 31;
  const int half = lane >> 4, lm = lane & 15;
  for (int nt = 0; nt < 2; ++nt) {
    int n0 = (wave * 2 + nt) * 16;
    int n = n0 + lm;
    v8f c = {};
    for (int k0 = 0; k0 < D; k0 += 4) {
      int kA = k0 + 2 * half;
      v2f a, b;
      a.x = sA[lm * lda + kA];
      a.y = sA[lm * lda + kA + 1];
      b.x = Wa[n * D + kA] - Wa[kA * D + n] - ((n == kA) ? GAMMAF : 0.0f);
      b.y = Wa[n * D + kA + 1] - Wa[(kA + 1) * D + n] -
            ((n == kA + 1) ? GAMMAF : 0.0f);
      c = __builtin_amdgcn_wmma_f32_16x16x4_f32(false, a, false, b, (short)0, c,
                                                false, false);
    }
    float bias = ba[n];
    for (int r = 0; r < 8; ++r) {
      int m = r0 + r + 8 * half;
      float pre = c[r] + phi[(size_t)m * D + n] + bias;
      float hv = sA[(r + 8 * half) * lda + n];
      hout[(size_t)m * D + n] = hv + EPSILONF * tanhf(pre);
    }
  }
}

// =====================================================================
extern "C" void kernel_launch(void* const* d_in, const int* in_sizes, int n_in,
                              void* d_out, int out_size, void* d_ws,
                              size_t ws_size, hipStream_t stream) {
  (void)in_sizes; (void)n_in; (void)out_size; (void)ws_size;
  const float* x = (const float*)d_in[0];
  const float* last_update = (const float*)d_in[1];
  const float* t = (const float*)d_in[2];
  const float* msg = (const float*)d_in[3];
  const int* edge_index = (const int*)d_in[4];
  const int* src = edge_index;            // edge_index[0, :]
  const int* dst = edge_index + NE;       // edge_index[1, :]
  const float* enc_W = (const float*)d_in[5];
  const float* enc_b = (const float*)d_in[6];
  const float* te_w = (const float*)d_in[7];
  const float* te_b = (const float*)d_in[8];
  const float* Wq = (const float*)d_in[9];
  const float* bq = (const float*)d_in[10];
  const float* Wk = (const float*)d_in[11];
  const float* bk = (const float*)d_in[12];
  const float* Wv = (const float*)d_in[13];
  const float* bv = (const float*)d_in[14];
  const float* We = (const float*)d_in[15];
  const float* be = (const float*)d_in[16];
  const float* Wa = (const float*)d_in[17];
  const float* ba = (const float*)d_in[18];

  char* ws = (char*)d_ws;
  size_t off = 0;
  auto alloc = [&](size_t bytes) -> char* {
    char* p = ws + off;
    off += (bytes + 255) & ~(size_t)255;
    return p;
  };
  float* h0 = (float*)alloc((size_t)NN * D * 4);
  float* h1 = (float*)alloc((size_t)NN * D * 4);
  float* q = (float*)alloc((size_t)NN * D * 4);
  float* k = (float*)alloc((size_t)NN * D * 4);
  float* v = (float*)alloc((size_t)NN * D * 4);
  float* phi = (float*)alloc((size_t)NN * D * 4);
  float* segsum = (float*)alloc((size_t)NN * 4);
  unsigned* segmax = (unsigned*)alloc((size_t)NN * 4);
  float* lg = (float*)alloc((size_t)NE * 4);
  float* efeat = (float*)alloc((size_t)NE * D * 4);

  // one-time precompute
  k_edgefeat<<<NE / 16, 128, 0, stream>>>(msg, t, last_update, src, te_w, te_b,
                                          We, be, efeat);
  k_encode<<<NN / 16, 128, 0, stream>>>(x, enc_W, enc_b, h0);

  float* hcur = h0;
  float* hnext = h1;
  const int edge_wave_blocks = (NE * 32 + 255) / 256;  // 1 wave per edge
  for (int it = 0; it < NUM_ITERS; ++it) {
    hipMemsetAsync(phi, 0, (size_t)NN * D * 4, stream);
    hipMemsetAsync(segsum, 0, (size_t)NN * 4, stream);
    hipMemsetAsync(segmax, 0, (size_t)NN * 4, stream);  // ordmap identity

    k_qkv<<<NN / 16, 128, 0, stream>>>(hcur, Wq, bq, Wk, bk, Wv, bv, q, k, v);
    k_logits<<<edge_wave_blocks, 256, 0, stream>>>(q, k, efeat, src, dst, lg,
                                                   segmax);
    k_exp<<<(NE + 255) / 256, 256, 0, stream>>>(lg, dst, segmax, segsum);
    k_scatter<<<edge_wave_blocks, 256, 0, stream>>>(v, efeat, src, dst, lg,
                                                    segsum, phi);
    float* out = (it == NUM_ITERS - 1) ? (float*)d_out : hnext;
    k_update<<<NN / 16, 128, 0, stream>>>(hcur, Wa, ba, phi, out);
    float* tmp = hcur;
    hcur = (it == NUM_ITERS - 1) ? hcur : hnext;
    hnext = tmp;
  }
}